// Model_9698036154926
// MI455X (gfx1250) — compile-verified
//
#include <hip/hip_runtime.h>

// Kalman-style linear scan, chunk-parallelized for MI455X (gfx1250, wave32).
//   x_{t+1} = M x_t + L y_t,  M = A - L*H ;  out_t = H x_t
// C=784 chunks of T=256 steps (padded past SEQ; padded outputs never stored).
// 16 chunks batched per wave => each step is a 64x64 @ 64x16 fp32 GEMM on
// V_WMMA_F32_16X16X4_F32. M/H A-fragments are register-resident; y is staged
// into LDS by the Tensor Data Mover (tensor_load_to_lds, double-buffered,
// s_wait_tensorcnt), keeping the cross-step critical path LDS+WMMA only.

#define NS   64
#define SEQ  200000
#define TT   256            // steps per chunk
#define CCH  784            // chunks (16*49); CCH*TT = 200704 >= SEQ
#define GRP  (CCH / 16)     // 49 wave-groups, 16 chunk-columns each
#define TB   8              // steps per TDM y-block
#define NTB  (TT / TB)      // 32 blocks per chunk

typedef __attribute__((ext_vector_type(2))) float v2f;
typedef __attribute__((ext_vector_type(8))) float v8f;
typedef __attribute__((ext_vector_type(4))) unsigned int v4u;
typedef __attribute__((ext_vector_type(8))) int v8i;
typedef __attribute__((ext_vector_type(4))) int v4i;

// workspace layout (float elements)
#define WS_MBAR   0                        // 64x64 row-major  (M = A - L*H)
#define WS_P      (NS * NS)                // 64x64 row-major  (M^TT)
#define WS_CARRY  (2 * NS * NS)            // [CCH][64]
#define WS_STARTS (2 * NS * NS + CCH*NS)   // [CCH][64]

#define WMMA_F32(a, b, c) \
  __builtin_amdgcn_wmma_f32_16x16x4_f32(false, (a), false, (b), (short)0, (c), false, false)

// ---- TDM: 2D tile (tile0 contiguous elems, tile1 rows of stride0) --------
// D# per CDNA5 ISA; data_size=4B; groups 2/3 zero.
// This toolchain's builtin is the 6-arg form:
//   (uint32x4 g0, int32x8 g1, int32x4, int32x4, int32x8, i32 cpol)
__device__ __forceinline__ void tdm_load_2d(unsigned lds_addr, const void* gptr,
                                            unsigned dim0, unsigned dim1,
                                            unsigned tile0, unsigned tile1,
                                            unsigned stride0) {
  unsigned long long ga = (unsigned long long)(size_t)gptr;
  v4u g0;
  g0[0] = 1u;                                   // count=1, user descriptor
  g0[1] = lds_addr;                             // LDS byte address [63:32]
  g0[2] = (unsigned)ga;                         // global addr [95:64]
  g0[3] = (unsigned)(ga >> 32) | (2u << 30);    // addr hi + type=2 ("image")
  v8i g1;
  g1[0] = (int)(2u << 16);                      // data_size=2 (4 bytes)
  g1[1] = (int)((dim0 & 0xFFFFu) << 16);        // tensor_dim0[15:0] @ [63:48]
  g1[2] = (int)((dim0 >> 16) | ((dim1 & 0xFFFFu) << 16)); // dim0 hi | dim1 lo
  g1[3] = (int)((dim1 >> 16) | (tile0 << 16));  // dim1 hi | tile_dim0 @[127:112]
  g1[4] = (int)tile1;                           // tile_dim1 @ [143:128]
  g1[5] = (int)stride0;                         // tensor_dim0_stride lo32
  g1[6] = 0;
  g1[7] = 0;
  v4i zz4 = {0, 0, 0, 0};
  v8i zz8 = {0, 0, 0, 0, 0, 0, 0, 0};
  __builtin_amdgcn_tensor_load_to_lds(g0, g1, zz4, zz4, zz8, 0);
}

// issue 16 column slabs y[0:64, tstart .. tstart+TB) -> Yt[col][64][TB]
__device__ __forceinline__ void issue_yblock(const float* __restrict__ y,
                                             int g, int tb, float* ytb) {
#pragma unroll 1
  for (int col = 0; col < 16; ++col) {
    long tstart = (long)(g * 16 + col) * TT + (long)tb * TB;
    if (tstart > (long)SEQ - TB) tstart = (long)SEQ - TB;  // stay in-bounds
    tdm_load_2d((unsigned)(size_t)&ytb[col * NS * TB], y + tstart,
                TB, NS, TB, NS, SEQ);
  }
}

// ---- 64x64 (row-major LDS) @ 64x16 (col-major LDS) += into 4 C/D tiles ---
__device__ __forceinline__ void mm64x16(const float* __restrict__ Ml,
                                        const float* __restrict__ Xl,
                                        v8f acc[4], int lm, int hi) {
#pragma unroll
  for (int ks = 0; ks < 16; ++ks) {
    const int k = ks * 4 + 2 * hi;
    v2f b;
    b.x = Xl[lm * NS + k];
    b.y = Xl[lm * NS + k + 1];
#pragma unroll
    for (int r = 0; r < 4; ++r) {
      const int m = r * 16 + lm;
      v2f a;
      a.x = Ml[m * NS + k];
      a.y = Ml[m * NS + k + 1];
      acc[r] = WMMA_F32(a, b, acc[r]);
    }
  }
}

// ---- kernel 0: Mbar = A - L*H --------------------------------------------
__global__ void k_prep(const float* __restrict__ A, const float* __restrict__ H,
                       const float* __restrict__ Lp, float* __restrict__ ws) {
  int i = blockIdx.x * blockDim.x + threadIdx.x;
  if (i < NS * NS) ws[WS_MBAR + i] = A[i] - Lp[0] * H[i];
}

// ---- kernel 1: P = Mbar^TT via repeated 64x64 GEMM (one wave, WMMA) ------
__global__ __launch_bounds__(32) void k_power(float* __restrict__ ws) {
  __shared__ float Ml[NS * NS];   // row-major
  __shared__ float Pl[NS * NS];   // col-major
  const int lane = threadIdx.x;
  for (int idx = lane; idx < NS * NS; idx += 32) {
    float m = ws[WS_MBAR + idx];
    Ml[idx] = m;
    Pl[(idx % NS) * NS + (idx / NS)] = m;   // Mbar^1
  }
  __syncthreads();
  const int lm = lane & 15, hi = lane >> 4;
  for (int t = 1; t < TT; ++t) {
#pragma unroll 1
    for (int cc = 0; cc < 4; ++cc) {        // in-place per 16-col block
      float* Xl = &Pl[cc * 16 * NS];
      v8f acc[4];
#pragma unroll
      for (int r = 0; r < 4; ++r) acc[r] = (v8f)0.0f;
      mm64x16(Ml, Xl, acc, lm, hi);
      __syncthreads();
#pragma unroll
      for (int r = 0; r < 4; ++r)
#pragma unroll
        for (int v = 0; v < 8; ++v)
          Xl[lm * NS + r * 16 + v + 8 * hi] = acc[r][v];
      __syncthreads();
    }
  }
  for (int idx = lane; idx < NS * NS; idx += 32)
    ws[WS_P + idx] = Pl[(idx % NS) * NS + (idx / NS)];
}

// ---- kernel 2 (phase 1): per-chunk carries from x=0 ----------------------
__global__ __launch_bounds__(32) void k_phase1(const float* __restrict__ y,
                                               const float* __restrict__ Lp,
                                               float* __restrict__ ws) {
  __shared__ float Ml[NS * NS];
  __shared__ float Xl[NS * 16];
  __shared__ float Yt[2 * 16 * NS * TB];   // double-buffered y slabs (64KB)
  const int lane = threadIdx.x, g = blockIdx.x;
  const int lm = lane & 15, hi = lane >> 4;

  tdm_load_2d((unsigned)(size_t)Ml, ws + WS_MBAR, NS, NS, NS, NS, NS);  // M
  issue_yblock(y, g, 0, &Yt[0]);                                        // y blk0
  for (int idx = lane; idx < NS * 16; idx += 32) Xl[idx] = 0.0f;
  const float L = Lp[0];
  __builtin_amdgcn_s_wait_tensorcnt((short)16);   // M (in-order) complete
  asm volatile("" ::: "memory");
  __syncthreads();

  v2f Mf[16][4];                                  // register-resident A frags
#pragma unroll
  for (int ks = 0; ks < 16; ++ks)
#pragma unroll
    for (int r = 0; r < 4; ++r) {
      const int k = ks * 4 + 2 * hi, m = r * 16 + lm;
      Mf[ks][r].x = Ml[m * NS + k];
      Mf[ks][r].y = Ml[m * NS + k + 1];
    }

  for (int tb = 0; tb < NTB; ++tb) {
    const float* ytb = &Yt[(tb & 1) * 16 * NS * TB];
    if (tb + 1 < NTB) {
      issue_yblock(y, g, tb + 1, &Yt[((tb + 1) & 1) * 16 * NS * TB]);
      __builtin_amdgcn_s_wait_tensorcnt((short)16);   // current block ready
    } else {
      __builtin_amdgcn_s_wait_tensorcnt((short)0);
    }
    asm volatile("" ::: "memory");
    __syncthreads();
#pragma unroll 1
    for (int tl = 0; tl < TB; ++tl) {
      v8f acc[4], acc2[4];
#pragma unroll
      for (int r = 0; r < 4; ++r) {
        acc2[r] = (v8f)0.0f;
#pragma unroll
        for (int v = 0; v < 8; ++v) {         // C := b_t = L*y (from LDS slab)
          const int m = r * 16 + v + 8 * hi;
          acc[r][v] = L * ytb[(lm * NS + m) * TB + tl];
        }
      }
#pragma unroll
      for (int ks = 0; ks < 16; ks += 2) {    // split-K: two 8-deep chains
        const int k0 = ks * 4 + 2 * hi, k1 = k0 + 4;
        v2f b0, b1;
        b0.x = Xl[lm * NS + k0];  b0.y = Xl[lm * NS + k0 + 1];
        b1.x = Xl[lm * NS + k1];  b1.y = Xl[lm * NS + k1 + 1];
#pragma unroll
        for (int r = 0; r < 4; ++r) acc[r] = WMMA_F32(Mf[ks][r], b0, acc[r]);
#pragma unroll
        for (int r = 0; r < 4; ++r) acc2[r] = WMMA_F32(Mf[ks + 1][r], b1, acc2[r]);
      }
      __syncthreads();
#pragma unroll
      for (int r = 0; r < 4; ++r)
#pragma unroll
        for (int v = 0; v < 8; ++v)
          Xl[lm * NS + r * 16 + v + 8 * hi] = acc[r][v] + acc2[r][v];
      __syncthreads();
    }
  }
  for (int idx = lane; idx < NS * 16; idx += 32)
    ws[WS_CARRY + g * NS * 16 + idx] = Xl[idx];   // carry[chunk][m]
}

// ---- kernel 3 (phase 2): CCH sequential chunk steps, VALU matvec ---------
__global__ __launch_bounds__(64) void k_phase2(float* __restrict__ ws) {
  __shared__ float Pl[NS * NS];
  __shared__ float xl[NS];
  const int tid = threadIdx.x;
  for (int idx = tid; idx < NS * NS; idx += 64) Pl[idx] = ws[WS_P + idx];
  xl[tid] = 0.0f;
  __syncthreads();
  for (int c = 0; c < CCH; ++c) {
    ws[WS_STARTS + c * NS + tid] = xl[tid];       // state entering chunk c
    float s = ws[WS_CARRY + c * NS + tid];
#pragma unroll 8
    for (int k = 0; k < NS; ++k) s += Pl[tid * NS + k] * xl[k];
    __syncthreads();
    xl[tid] = s;                                  // x_{c+1} = P x_c + carry_c
    __syncthreads();
  }
}

// ---- kernel 4 (phase 3): replay from true starts, emit H@x ---------------
__global__ __launch_bounds__(32) void k_phase3(const float* __restrict__ y,
                                               const float* __restrict__ H,
                                               const float* __restrict__ Lp,
                                               const float* __restrict__ ws,
                                               float* __restrict__ out) {
  __shared__ float Ml[NS * NS];
  __shared__ float Hl[NS * NS];
  __shared__ float Xl[NS * 16];
  __shared__ float Yt[2 * 16 * NS * TB];
  const int lane = threadIdx.x, g = blockIdx.x;
  const int lm = lane & 15, hi = lane >> 4;

  tdm_load_2d((unsigned)(size_t)Ml, ws + WS_MBAR, NS, NS, NS, NS, NS);
  tdm_load_2d((unsigned)(size_t)Hl, H, NS, NS, NS, NS, NS);
  issue_yblock(y, g, 0, &Yt[0]);
  for (int idx = lane; idx < NS * 16; idx += 32)
    Xl[idx] = ws[WS_STARTS + g * NS * 16 + idx];
  const float L = Lp[0];
  __builtin_amdgcn_s_wait_tensorcnt((short)16);   // M and H complete
  asm volatile("" ::: "memory");
  __syncthreads();

  v2f Hf[16][4];                                  // H register-resident
#pragma unroll
  for (int ks = 0; ks < 16; ++ks)
#pragma unroll
    for (int r = 0; r < 4; ++r) {
      const int k = ks * 4 + 2 * hi, m = r * 16 + lm;
      Hf[ks][r].x = Hl[m * NS + k];
      Hf[ks][r].y = Hl[m * NS + k + 1];
    }

  for (int tb = 0; tb < NTB; ++tb) {
    const float* ytb = &Yt[(tb & 1) * 16 * NS * TB];
    if (tb + 1 < NTB) {
      issue_yblock(y, g, tb + 1, &Yt[((tb + 1) & 1) * 16 * NS * TB]);
      __builtin_amdgcn_s_wait_tensorcnt((short)16);
    } else {
      __builtin_amdgcn_s_wait_tensorcnt((short)0);
    }
    asm volatile("" ::: "memory");
    __syncthreads();
#pragma unroll 1
    for (int tl = 0; tl < TB; ++tl) {
      v8f yh[4], acc[4];
#pragma unroll
      for (int r = 0; r < 4; ++r) {
        yh[r] = (v8f)0.0f;
#pragma unroll
        for (int v = 0; v < 8; ++v) {
          const int m = r * 16 + v + 8 * hi;
          acc[r][v] = L * ytb[(lm * NS + m) * TB + tl];
        }
      }
#pragma unroll
      for (int ks = 0; ks < 16; ++ks) {       // shared B frag for both GEMMs
        const int k = ks * 4 + 2 * hi;
        v2f b;
        b.x = Xl[lm * NS + k];  b.y = Xl[lm * NS + k + 1];
#pragma unroll
        for (int r = 0; r < 4; ++r) {
          v2f a;                               // M from LDS (off critical path)
          a.x = Ml[(r * 16 + lm) * NS + k];
          a.y = Ml[(r * 16 + lm) * NS + k + 1];
          acc[r] = WMMA_F32(a, b, acc[r]);     // x_{t+1} partial
          yh[r]  = WMMA_F32(Hf[ks][r], b, yh[r]);  // out_t partial
        }
      }
      const long tg = (long)(g * 16 + lm) * TT + (long)(tb * TB + tl);
      if (tg < SEQ) {                          // padded tail never stored
#pragma unroll
        for (int r = 0; r < 4; ++r)
#pragma unroll
          for (int v = 0; v < 8; ++v)
            out[(long)(r * 16 + v + 8 * hi) * SEQ + tg] = yh[r][v];
      }
      __syncthreads();
#pragma unroll
      for (int r = 0; r < 4; ++r)
#pragma unroll
        for (int v = 0; v < 8; ++v)
          Xl[lm * NS + r * 16 + v + 8 * hi] = acc[r][v];
      __syncthreads();
    }
  }
}

extern "C" void kernel_launch(void* const* d_in, const int* in_sizes, int n_in,
                              void* d_out, int out_size, void* d_ws, size_t ws_size,
                              hipStream_t stream) {
  const float* y = (const float*)d_in[0];   // [64, 200000]
  const float* A = (const float*)d_in[1];   // [64, 64]
  const float* H = (const float*)d_in[2];   // [64, 64]
  const float* L = (const float*)d_in[3];   // scalar
  float* out = (float*)d_out;               // [64, 200000]
  float* ws  = (float*)d_ws;                // 2*4096 + 2*784*64 floats = 434 KB

  k_prep  <<<(NS * NS + 255) / 256, 256, 0, stream>>>(A, H, L, ws);
  k_power <<<1,   32, 0, stream>>>(ws);
  k_phase1<<<GRP, 32, 0, stream>>>(y, L, ws);
  k_phase2<<<1,   64, 0, stream>>>(ws);
  k_phase3<<<GRP, 32, 0, stream>>>(y, H, L, ws, out);
}